// LanczosResamplingRenderer_73469710565398
// MI455X (gfx1250) — compile-verified
//
#include <hip/hip_runtime.h>
#include <hip/hip_bf16.h>

// ---------------------------------------------------------------------------
// LanczosResamplingRenderer for MI455X (gfx1250, wave32)
//
// FFT convolution implemented as a tensor-core four-step FFT:
//   1024 = 16 x 64 :  GEMM(DFT16) -> twiddle -> GEMM(DFT64)
// using V_WMMA_F32_16X16X4_F32 with all data resident in LDS.
// ---------------------------------------------------------------------------

typedef __attribute__((ext_vector_type(2))) float v2f;
typedef __attribute__((ext_vector_type(8))) float v8f;

#define FN        1024
#define HALF_F    512
#define NCH       3
#define TBLF      10752   // d16r(256) d16i(256) d64r(4096) d64i(4096) twr(1024) twi(1024)
#define LDSF      (4096 + TBLF)

// ---------------------------------------------------------------------------
// Table init: DFT16[k][n]=W16^{kn}, DFT64, twiddle W1024^{k1 n2}.
// Angles reduced mod N before sin/cos for accuracy.
// ---------------------------------------------------------------------------
__global__ void init_tables_kernel(float* __restrict__ tbl) {
  int i = blockIdx.x * blockDim.x + threadIdx.x;
  const float TWO_PI = 6.28318530717958647692f;
  float* d16r = tbl;
  float* d16i = tbl + 256;
  float* d64r = tbl + 512;
  float* d64i = tbl + 4608;
  float* twr  = tbl + 8704;
  float* twi  = tbl + 9728;
  if (i < 256) {
    int r = i >> 4, c = i & 15;
    int m = (r * c) & 15;
    float a = -TWO_PI * (float)m / 16.0f;
    d16r[i] = cosf(a); d16i[i] = sinf(a);
  }
  if (i < 4096) {
    int r = i >> 6, c = i & 63;
    int m = (r * c) & 63;
    float a = -TWO_PI * (float)m / 64.0f;
    d64r[i] = cosf(a); d64i[i] = sinf(a);
  }
  if (i < 1024) {
    int k1 = i >> 6, n2 = i & 63;
    int m = (k1 * n2) & 1023;
    float a = -TWO_PI * (float)m / 1024.0f;
    twr[i] = cosf(a); twi[i] = sinf(a);
  }
}

// ---------------------------------------------------------------------------
// Per-wave complex GEMM tile: C(16x16) += A(16xK) * B(Kx16) via WMMA f32.
// A row-major (lda), B row-major (ldb), B column tile starts at bn0.
// Complex: Cr += Ar*Br - Ai*Bi ; Ci += Ar*Bi + Ai*Br  (4 real WMMA chains).
// Layouts per CDNA5 ISA 7.12.2:
//   A 16x4 tile : lane<16 -> K=0,1 ; lane>=16 -> K=2,3 ; row = lane%16
//   B 4x16 tile : VGPR j -> K=j (lanes 0-15), K=j+2 (lanes 16-31); N = lane%16
//   C/D 16x16   : VGPR r -> M=r (lanes 0-15), M=r+8 (lanes 16-31); N = lane%16
// ---------------------------------------------------------------------------
__device__ __forceinline__ void cgemm16(
    const float* __restrict__ Ar, const float* __restrict__ Ai, int lda,
    const float* __restrict__ Br, const float* __restrict__ Bi, int ldb,
    int K, int bn0, int lrow, int hp, v8f& cr, v8f& ci)
{
  for (int c4 = 0; c4 < K; c4 += 4) {
    const int ka = c4 + hp * 2;
    v2f ar, ai, br, bi;
    ar.x = Ar[lrow * lda + ka];     ar.y = Ar[lrow * lda + ka + 1];
    ai.x = Ai[lrow * lda + ka];     ai.y = Ai[lrow * lda + ka + 1];
    br.x = Br[(ka    ) * ldb + bn0 + lrow];
    br.y = Br[(ka + 1) * ldb + bn0 + lrow];
    bi.x = Bi[(ka    ) * ldb + bn0 + lrow];
    bi.y = Bi[(ka + 1) * ldb + bn0 + lrow];
    v2f nbi = -bi;   // f32 WMMA NEG only covers C; negate B in VALU
    cr = __builtin_amdgcn_wmma_f32_16x16x4_f32(false, ar, false, br,  (short)0, cr, false, false);
    cr = __builtin_amdgcn_wmma_f32_16x16x4_f32(false, ai, false, nbi, (short)0, cr, false, false);
    ci = __builtin_amdgcn_wmma_f32_16x16x4_f32(false, ar, false, bi,  (short)0, ci, false, false);
    ci = __builtin_amdgcn_wmma_f32_16x16x4_f32(false, ai, false, br,  (short)0, ci, false, false);
  }
}

// ---------------------------------------------------------------------------
// In-place forward DFT of 1024 complex values in LDS. blockDim.x == 128.
//   n = n2 + 64*n1, k = k1 + 16*k2
//   Step A: T[k1][n2]  = sum_n1 x[n2+64n1] * W16^{k1 n1}      (WMMA GEMM, K=16)
//   Step B: T'[k1][n2] = T[k1][n2] * W1024^{k1 n2}            (VALU, fused in store)
//   Step C: x[k1+16k2] = sum_n2 T'[k1][n2] * W64^{n2 k2}      (WMMA GEMM, K=64)
// ---------------------------------------------------------------------------
__device__ __forceinline__ void fft1024(
    float* __restrict__ xre, float* __restrict__ xim,
    float* __restrict__ tre, float* __restrict__ tim,
    const float* __restrict__ d16r, const float* __restrict__ d16i,
    const float* __restrict__ d64r, const float* __restrict__ d64i,
    const float* __restrict__ twr,  const float* __restrict__ twi)
{
  const int lane = threadIdx.x & 31;
  const int lrow = lane & 15;
  const int hp   = lane >> 4;
  const int n0   = (threadIdx.x >> 5) << 4;   // wave id * 16 : this wave's N tile
  __syncthreads();

  // Step A: Xmat[n1][n2] = x[n2 + 64*n1] is row-major with stride 64.
  v8f cr = {}, ci = {};
  cgemm16(d16r, d16i, 16, xre, xim, 64, 16, n0, lrow, hp, cr, ci);

  // Step B fused into D-tile store: lane element r holds T[r+8*hp][n0+lrow].
#pragma unroll
  for (int r = 0; r < 8; ++r) {
    const int k1 = r + hp * 8;
    const int n2 = n0 + lrow;
    const float wr = twr[k1 * 64 + n2], wi = twi[k1 * 64 + n2];
    const float a = cr[r], b = ci[r];
    tre[k1 * 64 + n2] = a * wr - b * wi;
    tim[k1 * 64 + n2] = a * wi + b * wr;
  }
  __syncthreads();

  // Step C: Out[k1][k2] = T'(16x64) x D64(64x64), this wave: k2 in [n0,n0+16)
  v8f orr = {}, oii = {};
  cgemm16(tre, tim, 64, d64r, d64i, 64, 64, n0, lrow, hp, orr, oii);
#pragma unroll
  for (int r = 0; r < 8; ++r) {
    const int k1 = r + hp * 8;
    const int k2 = n0 + lrow;
    xre[k1 + 16 * k2] = orr[r];
    xim[k1 + 16 * k2] = oii[r];
  }
  __syncthreads();
}

__device__ __forceinline__ void load_tables(float* __restrict__ t,
                                            const float* __restrict__ tbl) {
  for (int i = threadIdx.x; i < TBLF; i += 128) t[i] = tbl[i];
}

#define FFT_LDS_SETUP()                                   \
  __shared__ float s[LDSF];                               \
  float* xre = s;                                         \
  float* xim = s + 1024;                                  \
  float* tre = s + 2048;                                  \
  float* tim = s + 3072;                                  \
  float* t   = s + 4096;                                  \
  const float* d16r = t;                                  \
  const float* d16i = t + 256;                            \
  const float* d64r = t + 512;                            \
  const float* d64i = t + 4608;                           \
  const float* twr  = t + 8704;                           \
  const float* twi  = t + 9728;                           \
  load_tables(t, tbl);

// ---------------------------------------------------------------------------
// Pass 1: forward row FFT of zero-padded model. grid = NCH*512, block = 128.
// ---------------------------------------------------------------------------
__global__ void __launch_bounds__(128)
fft_rows_fwd_kernel(const float* __restrict__ model,
                    float* __restrict__ X1r, float* __restrict__ X1i,
                    const float* __restrict__ tbl)
{
  FFT_LDS_SETUP();
  const int row = blockIdx.x & 511;
  const int ch  = blockIdx.x >> 9;
  for (int i = threadIdx.x; i < FN; i += 128) {
    xre[i] = (i < HALF_F) ? model[(ch * 512 + row) * 512 + i] : 0.0f;
    xim[i] = 0.0f;
  }
  fft1024(xre, xim, tre, tim, d16r, d16i, d64r, d64i, twr, twi);
  for (int i = threadIdx.x; i < FN; i += 128) {
    X1r[(ch * 512 + row) * FN + i] = xre[i];
    X1i[(ch * 512 + row) * FN + i] = xim[i];
  }
}

// ---------------------------------------------------------------------------
// Pass 2 (fused): per column — fwd col FFT, Hermitian-extended kernel multiply
// (conjugate for inverse folded in), fwd FFT, conj + 1/N scale, write back
// in place (rows 0..511 only: crop of H). grid = NCH*1024, block = 128.
// ---------------------------------------------------------------------------
__global__ void __launch_bounds__(128)
fft_cols_conv_kernel(float* __restrict__ X1r, float* __restrict__ X1i,
                     const float* __restrict__ kr, const float* __restrict__ ki,
                     const float* __restrict__ tbl)
{
  FFT_LDS_SETUP();
  const int col = blockIdx.x & 1023;
  const int ch  = blockIdx.x >> 10;
  for (int i = threadIdx.x; i < FN; i += 128) {
    if (i < HALF_F) {
      xre[i] = X1r[(ch * 512 + i) * FN + col];
      xim[i] = X1i[(ch * 512 + i) * FN + col];
    } else { xre[i] = 0.0f; xim[i] = 0.0f; }
  }
  fft1024(xre, xim, tre, tim, d16r, d16i, d64r, d64i, twr, twi);

  // spectrum[y=i][x=col] *= Kfull ; then conjugate for the inverse transform
  for (int i = threadIdx.x; i < FN; i += 128) {
    float krv, kiv;
    if (col <= 512) {
      krv = kr[(ch * FN + i) * 513 + col];
      kiv = ki[(ch * FN + i) * 513 + col];
    } else {                       // Hermitian extension: conj(K[(F-y)%F][F-x])
      const int yy = (FN - i) & (FN - 1);
      const int xx = FN - col;
      krv =  kr[(ch * FN + yy) * 513 + xx];
      kiv = -ki[(ch * FN + yy) * 513 + xx];
    }
    const float a = xre[i], b = xim[i];
    xre[i] = a * krv - b * kiv;
    xim[i] = -(a * kiv + b * krv);  // conj(X*K) for IFFT-via-forward-FFT
  }
  fft1024(xre, xim, tre, tim, d16r, d16i, d64r, d64i, twr, twi);

  const float inv = 1.0f / (float)FN;
  for (int i = threadIdx.x; i < HALF_F; i += 128) {
    X1r[(ch * 512 + i) * FN + col] =  xre[i] * inv;
    X1i[(ch * 512 + i) * FN + col] = -xim[i] * inv;
  }
}

// ---------------------------------------------------------------------------
// Pass 3: inverse row FFT (conj -> fwd -> scale), real part, crop to 512.
// grid = NCH*512, block = 128.
// ---------------------------------------------------------------------------
__global__ void __launch_bounds__(128)
ifft_rows_real_kernel(const float* __restrict__ X1r, const float* __restrict__ X1i,
                      float* __restrict__ conv, const float* __restrict__ tbl)
{
  FFT_LDS_SETUP();
  const int row = blockIdx.x & 511;
  const int ch  = blockIdx.x >> 9;
  for (int i = threadIdx.x; i < FN; i += 128) {
    xre[i] =  X1r[(ch * 512 + row) * FN + i];
    xim[i] = -X1i[(ch * 512 + row) * FN + i];
  }
  fft1024(xre, xim, tre, tim, d16r, d16i, d64r, d64i, twr, twi);
  const float inv = 1.0f / (float)FN;
  for (int i = threadIdx.x; i < HALF_F; i += 128)
    conv[(ch * 512 + row) * 512 + i] = xre[i] * inv;   // real part of IFFT
}

// ---------------------------------------------------------------------------
// Pass 4: Lanczos-5 resample at warped coordinates. One thread per pixel.
// ---------------------------------------------------------------------------
__device__ __forceinline__ float sincf(float u) {
  const float pu = 3.14159265358979323846f * u;
  return (fabsf(u) < 1e-7f) ? 1.0f : __sinf(pu) / pu;
}
__device__ __forceinline__ float lanczos5(float u) {
  return (fabsf(u) < 5.0f) ? sincf(u) * sincf(u * 0.2f) : 0.0f;
}

__global__ void __launch_bounds__(256)
lanczos_resample_kernel(const float* __restrict__ conv,
                        const float* __restrict__ warp,
                        float* __restrict__ out)
{
  const int p = blockIdx.x * blockDim.x + threadIdx.x;
  if (p >= 512 * 512) return;
  const float y = warp[2 * p + 0];
  const float x = warp[2 * p + 1];
  const int yf = (int)floorf(y);
  const int xf = (int)floorf(x);

  float wy[10], wx[10];
  int   yi[10], xi[10];
#pragma unroll
  for (int tt = 0; tt < 10; ++tt) {
    const int iy = yf - 4 + tt;
    wy[tt] = (iy >= 0 && iy < 512) ? lanczos5(y - (float)iy) : 0.0f;
    yi[tt] = iy < 0 ? 0 : (iy > 511 ? 511 : iy);
    const int ix = xf - 4 + tt;
    wx[tt] = (ix >= 0 && ix < 512) ? lanczos5(x - (float)ix) : 0.0f;
    xi[tt] = ix < 0 ? 0 : (ix > 511 ? 511 : ix);
  }

  for (int c = 0; c < NCH; ++c) {
    const float* img = conv + c * 512 * 512;
    __builtin_prefetch(img + yi[0] * 512 + xi[0], 0, 0);  // global_prefetch_b8
    float acc = 0.0f;
#pragma unroll
    for (int i = 0; i < 10; ++i) {
      const float* rowp = img + yi[i] * 512;
      float rs = 0.0f;
#pragma unroll
      for (int j = 0; j < 10; ++j) rs += wx[j] * rowp[xi[j]];
      acc += wy[i] * rs;
    }
    out[c * 512 * 512 + p] = acc;   // SCALE^2 == 1
  }
}

// ---------------------------------------------------------------------------
// Launch. Inputs: model(3,512,512) f32, warp(512,512,2) f32,
// kernel_fft_real/imag (3,1024,513) f32. Output: (3,512,512) f32.
// Workspace: tables (42KB) + X1r/X1i (6MB each) + conv (3MB) ~= 15.1 MB.
// ---------------------------------------------------------------------------
extern "C" void kernel_launch(void* const* d_in, const int* in_sizes, int n_in,
                              void* d_out, int out_size, void* d_ws, size_t ws_size,
                              hipStream_t stream) {
  const float* model = (const float*)d_in[0];
  const float* warp  = (const float*)d_in[1];
  const float* kr    = (const float*)d_in[2];
  const float* ki    = (const float*)d_in[3];
  float* out = (float*)d_out;

  float* ws   = (float*)d_ws;
  float* tbl  = ws;                               // TBLF floats
  float* X1r  = tbl + TBLF;                       // 3*512*1024
  float* X1i  = X1r + NCH * 512 * FN;             // 3*512*1024
  float* conv = X1i + NCH * 512 * FN;             // 3*512*512

  init_tables_kernel<<<dim3((TBLF + 255) / 256), dim3(256), 0, stream>>>(tbl);
  fft_rows_fwd_kernel<<<dim3(NCH * 512), dim3(128), 0, stream>>>(model, X1r, X1i, tbl);
  fft_cols_conv_kernel<<<dim3(NCH * FN), dim3(128), 0, stream>>>(X1r, X1i, kr, ki, tbl);
  ifft_rows_real_kernel<<<dim3(NCH * 512), dim3(128), 0, stream>>>(X1r, X1i, conv, tbl);
  lanczos_resample_kernel<<<dim3((512 * 512 + 255) / 256), dim3(256), 0, stream>>>(conv, warp, out);
}